// ImageArm_64596308132122
// MI455X (gfx1250) — compile-verified
//
#include <hip/hip_runtime.h>
#include <math.h>

// MI455X / gfx1250, wave32. Memory-bound MLP (AI ~31 FLOP/B << ~31x what
// 23.3 TB/s could feed), so: stream X coalesced, weights in registers,
// per-16-row-tile GEMMs via V_WMMA_F32_16X16X4_F32, wave-private LDS for
// the D-layout -> A-layout transpose between layers.

typedef __attribute__((ext_vector_type(2))) float v2f;
typedef __attribute__((ext_vector_type(4))) float v4f;
typedef __attribute__((ext_vector_type(8))) float v8f;

#define WAVES_PER_BLOCK 8
#define TILE_ROWS 16
#define LSTRIDE 36   // 34 inputs padded to 36 (9 k-chunks of 4); conflict-free on 64 banks

__device__ __forceinline__ v8f wmma4(v2f a, v2f b, v8f c) {
  // D = A(16x4 f32) * B(4x16 f32) + C(16x16 f32)
  return __builtin_amdgcn_wmma_f32_16x16x4_f32(
      /*neg_a=*/false, a, /*neg_b=*/false, b,
      /*c_mod=*/(short)0, c, /*reuse_a=*/false, /*reuse_b=*/false);
}

__global__ void __launch_bounds__(256)
arm_mlp_wmma_kernel(const float* __restrict__ x,  const float* __restrict__ sp,
                    const float* __restrict__ W0, const float* __restrict__ b0,
                    const float* __restrict__ W1, const float* __restrict__ b1,
                    const float* __restrict__ W2, const float* __restrict__ b2,
                    float* __restrict__ out, int nTiles) {
  __shared__ __align__(16) float lds[WAVES_PER_BLOCK * 2 * TILE_ROWS * LSTRIDE];

  const int lane = threadIdx.x & 31;
  const int wave = threadIdx.x >> 5;
  float* buf0 = &lds[wave * (2 * TILE_ROWS * LSTRIDE)];
  float* buf1 = buf0 + TILE_ROWS * LSTRIDE;

  const int lo = lane & 15;  // N column (B/C/D) or row (A reads)
  const int hi = lane >> 4;  // lane-half selects K pair / +8 rows

  // ---- Preload weights into B-layout register chunks (lane lo = N, K = 4s + v + 2*hi) ----
  v2f w0c[2][9];
#pragma unroll
  for (int t = 0; t < 2; ++t)
#pragma unroll
    for (int s = 0; s < 9; ++s) {
      const int N = t * 16 + lo;
      const int K0 = 4 * s + 2 * hi;
      v2f w;
      w.x = (K0 + 0 < 34) ? W0[N * 34 + K0 + 0] : 0.0f;  // K=34,35 are zero-pad
      w.y = (K0 + 1 < 34) ? W0[N * 34 + K0 + 1] : 0.0f;
      w0c[t][s] = w;
    }
  v2f w1c[2][8];
#pragma unroll
  for (int t = 0; t < 2; ++t)
#pragma unroll
    for (int s = 0; s < 8; ++s) {
      const int N = t * 16 + lo;
      const int K0 = 4 * s + 2 * hi;
      v2f w; w.x = W1[N * 32 + K0]; w.y = W1[N * 32 + K0 + 1];
      w1c[t][s] = w;
    }
  v2f w2c[8];
#pragma unroll
  for (int s = 0; s < 8; ++s) {
    const int K0 = 4 * s + 2 * hi;
    v2f w;
    if (lo < 4) { w.x = W2[lo * 32 + K0]; w.y = W2[lo * 32 + K0 + 1]; }
    else        { w.x = 0.0f; w.y = 0.0f; }
    w2c[s] = w;
  }
  const float bias0[2] = {b0[lo], b0[16 + lo]};
  const float bias1[2] = {b1[lo], b1[16 + lo]};
  const float bias2    = (lo < 4) ? b2[lo] : 0.0f;

  const int waveGlobal = blockIdx.x * WAVES_PER_BLOCK + wave;
  const int waveTotal  = gridDim.x * WAVES_PER_BLOCK;

  for (int tile = waveGlobal; tile < nTiles; tile += waveTotal) {
    const int row0 = tile * TILE_ROWS;

    // ---- Stage X tile: 16x32 f32 is a contiguous 2KB block -> 4 coalesced float4/lane ----
    const v4f* xs = (const v4f*)(x + (size_t)row0 * 32);
#pragma unroll
    for (int j = 0; j < 4; ++j) {
      v4f v = xs[lane + 32 * j];
      const int f = (lane + 32 * j) * 4;          // flat float index in tile
      const int r = f >> 5, c = f & 31;
      *(v4f*)&buf0[r * LSTRIDE + c] = v;
    }
    // sp tile: 16x2 contiguous -> features 32,33
    buf0[(lane >> 1) * LSTRIDE + 32 + (lane & 1)] = sp[(size_t)row0 * 2 + lane];
    if (lane < 16) {                               // features 34,35 = zero pad
      v2f z; z.x = 0.0f; z.y = 0.0f;
      *(v2f*)&buf0[lane * LSTRIDE + 34] = z;
    }
    __builtin_amdgcn_wave_barrier();               // LDS is in-order per wave; just stop reordering

    // ---- Layer 0: h1 = relu(X @ W0^T + b0), K=36 in 9 WMMA steps, 2 N-tiles ----
    v8f h1[2];
#pragma unroll
    for (int t = 0; t < 2; ++t) {
      v8f acc = {};
#pragma unroll
      for (int s = 0; s < 9; ++s) {
        v2f a = *(const v2f*)&buf0[lo * LSTRIDE + 4 * s + 2 * hi];
        acc = wmma4(a, w0c[t][s], acc);
      }
#pragma unroll
      for (int k = 0; k < 8; ++k) h1[t][k] = fmaxf(acc[k] + bias0[t], 0.0f);
    }
    // D-layout -> row-major LDS (row = v + 8*hi, feat = t*16 + lo)
#pragma unroll
    for (int t = 0; t < 2; ++t)
#pragma unroll
      for (int v = 0; v < 8; ++v)
        buf1[(v + 8 * hi) * LSTRIDE + t * 16 + lo] = h1[t][v];
    __builtin_amdgcn_wave_barrier();

    // ---- Layer 1: h2 = relu(h1 @ W1^T + b1 + h1)  (residual add stays in registers) ----
    v8f h2[2];
#pragma unroll
    for (int t = 0; t < 2; ++t) {
      v8f acc = {};
#pragma unroll
      for (int s = 0; s < 8; ++s) {
        v2f a = *(const v2f*)&buf1[lo * LSTRIDE + 4 * s + 2 * hi];
        acc = wmma4(a, w1c[t][s], acc);
      }
#pragma unroll
      for (int k = 0; k < 8; ++k)
        h2[t][k] = fmaxf(acc[k] + bias1[t] + h1[t][k], 0.0f);
    }
    // h2 -> buf0 cols 0..31 (sp preserved in cols 32..33)
#pragma unroll
    for (int t = 0; t < 2; ++t)
#pragma unroll
      for (int v = 0; v < 8; ++v)
        buf0[(v + 8 * hi) * LSTRIDE + t * 16 + lo] = h2[t][v];
    __builtin_amdgcn_wave_barrier();

    // ---- Layer 2: raw = h2 @ W2^T + b2 (only N=0..3 valid) ----
    v8f acc = {};
#pragma unroll
    for (int s = 0; s < 8; ++s) {
      v2f a = *(const v2f*)&buf0[lo * LSTRIDE + 4 * s + 2 * hi];
      acc = wmma4(a, w2c[s], acc);
    }
    if (lo < 4) {
#pragma unroll
      for (int v = 0; v < 8; ++v)
        buf1[(v + 8 * hi) * LSTRIDE + lo] = acc[v] + bias2;
    }
    __builtin_amdgcn_wave_barrier();

    // ---- Epilogue: out = sp + p * sigmoid(gate); one coalesced 128B store per wave ----
    {
      const int r = lane >> 1, n = lane & 1;
      const float p = buf1[r * LSTRIDE + n];
      const float g = buf1[r * LSTRIDE + n + 2];
      const float s = buf0[r * LSTRIDE + 32 + n];
      const float sig = 1.0f / (1.0f + __expf(-g));
      out[(size_t)row0 * 2 + lane] = fmaf(p, sig, s);
    }
    __builtin_amdgcn_wave_barrier();  // next iter's stores must not pass these loads
  }
}

extern "C" void kernel_launch(void* const* d_in, const int* in_sizes, int n_in,
                              void* d_out, int out_size, void* d_ws, size_t ws_size,
                              hipStream_t stream) {
  const float* x  = (const float*)d_in[0];
  const float* sp = (const float*)d_in[1];
  const float* W0 = (const float*)d_in[2];
  const float* b0 = (const float*)d_in[3];
  const float* W1 = (const float*)d_in[4];
  const float* b1 = (const float*)d_in[5];
  const float* W2 = (const float*)d_in[6];
  const float* b2 = (const float*)d_in[7];
  float* out = (float*)d_out;

  const int B = in_sizes[0] / 32;          // 2097152
  const int nTiles = B / TILE_ROWS;        // 131072 (B divisible by 16)
  int blocks = (nTiles + (WAVES_PER_BLOCK * 8) - 1) / (WAVES_PER_BLOCK * 8);  // ~8 tiles/wave
  if (blocks < 1) blocks = 1;
  if (blocks > 8192) blocks = 8192;

  arm_mlp_wmma_kernel<<<blocks, 256, 0, stream>>>(x, sp, W0, b0, W1, b1, W2, b2,
                                                  out, nTiles);
}